// GNP_Encoder_16561393893850
// MI455X (gfx1250) — compile-verified
//
#include <hip/hip_runtime.h>
#include <hip/hip_bf16.h>
#include <math.h>

typedef float v2f __attribute__((ext_vector_type(2)));
typedef float v8f __attribute__((ext_vector_type(8)));

#define N_FEAT1 256
#define N_HID   128
#define N_OUT2  64

// ---------------------------------------------------------------------------
// GEMM via V_WMMA_F32_16X16X4_F32 (fp32 matrix cores, exact fp32 accumulate).
// C[M x 128] = op(A)[M x K] @ B[K x 128],  ld(A)=K, ld(B)=ld(C)=128.
// One wave computes a full 16-row x 128-col strip: 8 v8f accumulators,
// A loaded once per k-step (float2), reused across all 8 N-tiles.
// Wave32 A-layout (16x4):  lane L holds row (L&15), K = (L>>4)*2 + {0,1}
// Wave32 B-layout (4x16):  lane L holds col (L&15), K = (L>>4)*2 + {0,1}
// Wave32 D-layout (16x16): VGPR r -> row r + 8*(L>>4), col (L&15)
// ---------------------------------------------------------------------------
template <int K, bool RELU>
__global__ void gemm_wmma_f32(const float* __restrict__ A,
                              const float* __restrict__ B,
                              float* __restrict__ C, int M) {
  const int wave = blockIdx.x * (blockDim.x >> 5) + (threadIdx.x >> 5);
  const int lane = threadIdx.x & 31;
  const int row0 = wave * 16;
  if (row0 >= M) return;                 // wave-uniform: EXEC all-1s for WMMA

  const int mlane = lane & 15;           // A-row / B-col / D-col within tile
  const int khalf = lane >> 4;           // 0 or 1: which K-pair this lane owns

  v8f acc[8] = {};

  const float* arow = A + (size_t)(row0 + mlane) * K + khalf * 2;

  for (int k = 0; k < K; k += 4) {
    float2 av = *(const float2*)(arow + k);       // A[m][k+2*khalf .. +1]
    if (RELU) { av.x = fmaxf(av.x, 0.0f); av.y = fmaxf(av.y, 0.0f); }
    v2f a; a.x = av.x; a.y = av.y;

    const float* bp = B + (size_t)(k + khalf * 2) * 128 + mlane;
#pragma unroll
    for (int t = 0; t < 8; ++t) {
      v2f b;
      b.x = bp[t * 16];                           // B[k+2*khalf  ][16t + n]
      b.y = bp[t * 16 + 128];                     // B[k+2*khalf+1][16t + n]
      acc[t] = __builtin_amdgcn_wmma_f32_16x16x4_f32(
          /*neg_a=*/false, a, /*neg_b=*/false, b,
          /*c_mod=*/(short)0, acc[t], /*reuse_a=*/false, /*reuse_b=*/false);
    }
  }

  const int mstore = row0 + (khalf << 3);         // rows r+0..7 or r+8..15
#pragma unroll
  for (int t = 0; t < 8; ++t) {
#pragma unroll
    for (int r = 0; r < 8; ++r) {
      C[(size_t)(mstore + r) * 128 + t * 16 + mlane] = acc[t][r];
    }
  }
}

// Wcat[k][0:64] = W2[k][:], Wcat[k][64:128] = W3[k][:]
__global__ void concat_w(const float* __restrict__ W2,
                         const float* __restrict__ W3,
                         float* __restrict__ Wcat) {
  int i = blockIdx.x * blockDim.x + threadIdx.x;  // 128*128 threads
  int k = i >> 7, d = i & 127;
  Wcat[i] = (d < 64) ? W2[k * 64 + d] : W3[k * 64 + (d - 64)];
}

// Layer-1 spmm: H[row[e], :128] += val[e] * S1[col[e], :128].
// One wave per edge, 4 floats per lane; gathered rows are L2-resident (51MB).
__global__ void spmm1(const float* __restrict__ S1,
                      const int* __restrict__ erow, const int* __restrict__ ecol,
                      const float* __restrict__ eval,
                      float* __restrict__ H, int nEdges) {
  int gid = blockIdx.x * blockDim.x + threadIdx.x;
  int e = gid >> 5;
  if (e >= nEdges) return;
  int lane = gid & 31;
  int r = erow[e], c = ecol[e];
  float v = eval[e];
  float4 s = *(const float4*)(S1 + (size_t)c * 128 + lane * 4);
  float* h = H + (size_t)r * 128 + lane * 4;
  atomicAdd(h + 0, v * s.x);
  atomicAdd(h + 1, v * s.y);
  atomicAdd(h + 2, v * s.z);
  atomicAdd(h + 3, v * s.w);
}

// rowsum2[j] = sum_d Scat[j][0:64]  (= sum over support2 row, for z_mu shortcut)
__global__ void rowsum_s2(const float* __restrict__ Scat,
                          float* __restrict__ rowsum, int M) {
  int gid = blockIdx.x * blockDim.x + threadIdx.x;
  int row = gid >> 5;
  if (row >= M) return;
  int lane = gid & 31;
  float2 s = *(const float2*)(Scat + (size_t)row * 128 + lane * 2);
  float v = s.x + s.y;
#pragma unroll
  for (int off = 16; off; off >>= 1) v += __shfl_xor(v, off, 32);
  if (lane == 0) rowsum[row] = v;
}

// Layer-2 spmm (logvar only) + z_mu edge reduction.
//   LV[row[e], :64] += val[e] * Scat[col[e], 64:128]
//   mu_sum         += val[e] * rowsum2[col[e]]
__global__ void spmm2(const float* __restrict__ Scat,
                      const float* __restrict__ rowsum2,
                      const int* __restrict__ erow, const int* __restrict__ ecol,
                      const float* __restrict__ eval,
                      float* __restrict__ LV, float* __restrict__ mu_sum,
                      int nEdges) {
  __shared__ float smem[8];               // blockDim=256 -> 8 waves
  int gid = blockIdx.x * blockDim.x + threadIdx.x;
  int e = gid >> 5;
  int lane = gid & 31;
  float mupart = 0.0f;
  if (e < nEdges) {
    int r = erow[e], c = ecol[e];
    float v = eval[e];
    float2 s = *(const float2*)(Scat + (size_t)c * 128 + 64 + lane * 2);
    float* lv = LV + (size_t)r * 64 + lane * 2;
    atomicAdd(lv + 0, v * s.x);
    atomicAdd(lv + 1, v * s.y);
    if (lane == 0) mupart = v * rowsum2[c];
  }
  int w = threadIdx.x >> 5;
  if (lane == 0) smem[w] = mupart;
  __syncthreads();
  if (threadIdx.x == 0) {
    float t = 0.0f;
    for (int i = 0; i < 8; ++i) t += smem[i];
    atomicAdd(mu_sum, t);
  }
}

// sum of exp(LV) over all M*64 elements
__global__ void exp_sum(const float* __restrict__ LV, float* __restrict__ out,
                        size_t n) {
  size_t stride = (size_t)gridDim.x * blockDim.x;
  float v = 0.0f;
  for (size_t i = blockIdx.x * (size_t)blockDim.x + threadIdx.x; i < n; i += stride)
    v += expf(LV[i]);
#pragma unroll
  for (int off = 16; off; off >>= 1) v += __shfl_xor(v, off, 32);
  __shared__ float smem[8];
  int lane = threadIdx.x & 31, w = threadIdx.x >> 5;
  if (lane == 0) smem[w] = v;
  __syncthreads();
  if (threadIdx.x == 0) {
    float t = 0.0f;
    for (int i = 0; i < 8; ++i) t += smem[i];
    atomicAdd(out, t);
  }
}

__global__ void finalize(const float* __restrict__ scal, float* __restrict__ out,
                         float inv_count) {
  out[0] = scal[0] * inv_count;           // z_mu
  out[1] = logf(scal[1] * inv_count);     // z_logvar
}

extern "C" void kernel_launch(void* const* d_in, const int* in_sizes, int n_in,
                              void* d_out, int out_size, void* d_ws, size_t ws_size,
                              hipStream_t stream) {
  const float* x    = (const float*)d_in[0];
  const int*   erow = (const int*)d_in[1];
  const int*   ecol = (const int*)d_in[2];
  const float* eval = (const float*)d_in[3];
  const float* W1   = (const float*)d_in[4];
  const float* W2   = (const float*)d_in[5];
  const float* W3   = (const float*)d_in[6];
  float* out = (float*)d_out;

  const int M = in_sizes[0] / N_FEAT1;    // 100000
  const int E = in_sizes[1];              // 1600000

  // workspace layout (S1 buffer reused as Scat after spmm1):
  char* ws = (char*)d_ws;
  const size_t bytes128 = (size_t)M * 128 * sizeof(float);   // 51.2 MB
  const size_t bytes64  = (size_t)M * 64  * sizeof(float);   // 25.6 MB
  float* S1      = (float*)ws;                               // then Scat
  float* H       = (float*)(ws + bytes128);
  float* LV      = (float*)(ws + 2 * bytes128);
  float* rowsum2 = (float*)(ws + 2 * bytes128 + bytes64);
  float* Wcat    = (float*)(ws + 2 * bytes128 + bytes64 + (size_t)M * sizeof(float));
  float* scal    = (float*)((char*)Wcat + 128 * 128 * sizeof(float)); // [mu_sum, exp_sum]

  // zero accumulators (fresh every call; graph-capture-safe)
  hipMemsetAsync(H, 0, bytes128, stream);
  hipMemsetAsync(LV, 0, bytes64, stream);
  hipMemsetAsync(scal, 0, 2 * sizeof(float), stream);

  // Wcat = [W2 | W3]
  concat_w<<<128, 128, 0, stream>>>(W2, W3, Wcat);

  // layer 1: support1 = x @ W1
  const int waves = (M + 15) / 16;
  const int gblocks = (waves + 3) / 4;    // 4 waves per 128-thread block
  gemm_wmma_f32<N_FEAT1, false><<<gblocks, 128, 0, stream>>>(x, W1, S1, M);

  // hidden1(pre-relu) = spmm(adj, support1)
  spmm1<<<(E * 32 + 255) / 256, 256, 0, stream>>>(S1, erow, ecol, eval, H, E);

  // layer 2 (relu fused into A loads): [support2|support3] = relu(H) @ [W2|W3]
  gemm_wmma_f32<N_HID, true><<<gblocks, 128, 0, stream>>>(H, Wcat, S1, M);

  // z_mu shortcut: per-node rowsum of support2
  rowsum_s2<<<(M * 32 + 255) / 256, 256, 0, stream>>>(S1, rowsum2, M);

  // logvar spmm + mu edge-sum
  spmm2<<<(E * 32 + 255) / 256, 256, 0, stream>>>(S1, rowsum2, erow, ecol, eval,
                                                  LV, &scal[0], E);

  // sum exp(logvar)
  exp_sum<<<2048, 256, 0, stream>>>(LV, &scal[1], (size_t)M * 64);

  finalize<<<1, 1, 0, stream>>>(scal, out, 1.0f / ((float)M * 64.0f));
}